// GeometricField_88029649699078
// MI455X (gfx1250) — compile-verified
//
#include <hip/hip_runtime.h>
#include <math.h>

#define GN 8192
#define TILE 16
#define THREADS 256
#define NWAVES 8
#define JSLICE (GN / NWAVES)   // 1024 columns per wave in pass 1

#define TWO_PI_F 6.2831853071795864769f
// -LAMBDA * log2(e) = -2.5 * 1.4426950408889634
#define NEG_LAMBDA_LOG2E (-3.6067376022224085f)

typedef __attribute__((ext_vector_type(2))) float v2f;
typedef __attribute__((ext_vector_type(8))) float v8f;

__global__ __launch_bounds__(THREADS) void GeometricField_kernel(
    const float* __restrict__ theta, const float* __restrict__ tau,
    float* __restrict__ theta_out, float* __restrict__ tau_out,
    float* __restrict__ attn)
{
    extern __shared__ float smem[];
    float* s_theta = smem;                 // GN
    float* s_tau   = s_theta + GN;         // GN
    float* s_cos   = s_tau   + GN;         // GN
    float* s_sin   = s_cos   + GN;         // GN
    float* s_acc   = s_sin   + GN;         // 16*16 WMMA tile accumulator
    float* s_rep   = s_acc   + 256;        // 16*2 repulsion sums
    float* s_inv   = s_rep   + 32;         // 16 per-row 1/rowsum

    const int tid   = threadIdx.x;
    const int tile0 = blockIdx.x * TILE;

    // ---- Stage inputs + trig into LDS (64KB inputs -> 128KB LDS) ----
    for (int idx = tid; idx < GN; idx += THREADS) {
        float th = theta[idx];
        s_theta[idx] = th;
        s_tau[idx]   = tau[idx];
        float sn, cs;
        __sincosf(th, &sn, &cs);
        s_cos[idx] = cs;
        s_sin[idx] = sn;
    }
    s_acc[tid] = 0.0f;                     // 256 entries == THREADS
    if (tid < 32) s_rep[tid] = 0.0f;
    __syncthreads();

    const int lane = tid & 31;
    const int wave = tid >> 5;
    const int n    = lane & 15;            // A-row within tile AND B-column
    const int kb   = (lane >> 4) * 2;      // this half-wave's K offsets {0,1} or {2,3}

    const int   i   = tile0 + n;           // global row this lane scores
    const float thi = s_theta[i];
    const float tui = s_tau[i];
    const float ci  = s_cos[i];
    const float si  = s_sin[i];

    // B operand source: column 0 -> cos_j, 1 -> sin_j, 2 -> tau_j, 3 -> 1.0, else 0
    const float* bsrc = (n == 1) ? s_sin : (n == 2) ? s_tau : s_cos;

    // ---- Pass 1: fused scores + repulsion; WMMA accumulates
    //      D[:,0]=Σs·cos  D[:,1]=Σs·sin  D[:,2]=Σs·tau  D[:,3]=Σs (rowsum) ----
    v8f  c   = {};
    float rpx = 0.0f, rpy = 0.0f;

    const int j0 = wave * JSLICE;
    const int j1 = j0 + JSLICE;
    for (int jc = j0; jc < j1; jc += 4) {
        const int j = jc + kb;             // this lane's two columns: j, j+1
        v2f a;
#pragma unroll
        for (int u = 0; u < 2; ++u) {
            const int jj  = j + u;
            const float thj  = s_theta[jj];
            const float diff = fabsf(thi - thj);
            const float dth  = fminf(diff, TWO_PI_F - diff);
            const float dist = fmaf(0.3f, fabsf(tui - s_tau[jj]), dth);
            float s = __builtin_amdgcn_exp2f(dist * NEG_LAMBDA_LOG2E);
            // repulsion: 0.15/(d^2) / (|p_i - p_j| + 1e-8)
            const float d  = dth + 0.1f;
            const float dx = ci - s_cos[jj];
            const float dy = si - s_sin[jj];
            const float sq = fmaf(dx, dx, dy * dy);
            const float nr = __builtin_amdgcn_sqrtf(sq) + 1e-8f;
            float w = 0.15f * __builtin_amdgcn_rcpf(d * d * nr);
            if (jj == i) { s = 0.0f; w = 0.0f; }     // diagonal excluded
            rpx = fmaf(w, dx, rpx);
            rpy = fmaf(w, dy, rpy);
            a[u] = s;
        }
        // Build B (4x16): row K = kb+u, column n; columns >3 are zero.
        const v2f bv = *(const v2f*)(bsrc + jc + kb);
        v2f b;
        b.x = (n < 3) ? bv.x : ((n == 3) ? 1.0f : 0.0f);
        b.y = (n < 3) ? bv.y : ((n == 3) ? 1.0f : 0.0f);
        c = __builtin_amdgcn_wmma_f32_16x16x4_f32(
                /*neg_a=*/false, a, /*neg_b=*/false, b,
                /*c_mod=*/(short)0, c, /*reuse_a=*/false, /*reuse_b=*/false);
    }

    // Merge the 8 waves' partial D tiles + repulsion partials in LDS.
#pragma unroll
    for (int p = 0; p < 8; ++p) {
        const int m = p + ((lane >> 4) << 3);        // D layout: VGPR p, half-wave -> M
        atomicAdd(&s_acc[m * 16 + n], c[p]);
    }
    atomicAdd(&s_rep[n * 2 + 0], rpx);
    atomicAdd(&s_rep[n * 2 + 1], rpy);
    __syncthreads();

    // ---- Per-row finalize: atan2 + tau_out; stash 1/rowsum for pass 2 ----
    if (tid < TILE) {
        const int r  = tid;
        const float Sx = s_acc[r * 16 + 0];
        const float Sy = s_acc[r * 16 + 1];
        const float St = s_acc[r * 16 + 2];
        const float rs = s_acc[r * 16 + 3];
        const float inv = 1.0f / rs;                 // rs > 0 (8191 positive terms)
        const float fx = fmaf(Sx, inv, s_rep[r * 2 + 0]);
        const float fy = fmaf(Sy, inv, s_rep[r * 2 + 1]);
        theta_out[tile0 + r] = atan2f(fy, fx);
        tau_out[tile0 + r]   = St * inv;
        s_inv[r] = inv;
    }
    __syncthreads();

    // ---- Pass 2: recompute scores, scale by 1/rowsum, write attn rows
    //      (recompute beats a 2x268MB read/rewrite round-trip) ----
    for (int rr = wave; rr < TILE; rr += NWAVES) {
        const int   row = tile0 + rr;
        const float thr = s_theta[row];
        const float tur = s_tau[row];
        const float inv = s_inv[rr];
        float* out_row = attn + (size_t)row * GN;
        for (int jb = lane * 4; jb < GN; jb += 128) {
            float4 v;
            float* vp = &v.x;
#pragma unroll
            for (int u = 0; u < 4; ++u) {
                const int jj = jb + u;
                const float diff = fabsf(thr - s_theta[jj]);
                const float dth  = fminf(diff, TWO_PI_F - diff);
                const float dist = fmaf(0.3f, fabsf(tur - s_tau[jj]), dth);
                const float s = __builtin_amdgcn_exp2f(dist * NEG_LAMBDA_LOG2E);
                vp[u] = (jj == row) ? 0.0f : s * inv;
            }
            *(float4*)(out_row + jb) = v;            // coalesced 512B per wave
        }
    }
}

extern "C" void kernel_launch(void* const* d_in, const int* in_sizes, int n_in,
                              void* d_out, int out_size, void* d_ws, size_t ws_size,
                              hipStream_t stream) {
    const float* theta = (const float*)d_in[0];
    const float* tau   = (const float*)d_in[1];
    float* out       = (float*)d_out;
    float* theta_out = out;           // N
    float* tau_out   = out + GN;      // N
    float* attn      = out + 2 * GN;  // N*N

    const size_t smem_bytes = (size_t)(4 * GN + 256 + 32 + 16) * sizeof(float);
    GeometricField_kernel<<<GN / TILE, THREADS, smem_bytes, stream>>>(
        theta, tau, theta_out, tau_out, attn);
}